// SpatialConsistencyLoss_70944269796033
// MI455X (gfx1250) — compile-verified
//
#include <hip/hip_runtime.h>
#include <hip/hip_bf16.h>
#include <math.h>

// Patch-mean + masked loss for SpatialConsistencyLoss on gfx1250.
// Memory-bound: 256 MB fp32 streamed once -> ~11 us floor at 23.3 TB/s.
// Hot loop = clause of 8x global_load_b128 (non-temporal: stream > 192MB L2,
// zero reuse) + 16 chained v_wmma_f32_16x16x4_f32 (B = ones as an
// accumulating 64-float adder on the matrix pipe).

typedef __attribute__((ext_vector_type(2))) float v2f;
typedef __attribute__((ext_vector_type(4))) float v4f;
typedef __attribute__((ext_vector_type(8))) float v8f;

#define IMG_W 8192
#define LOGC  (-2.2999372975086f)   /* log(0.04 * sqrt(2*pi)) */

__global__ __launch_bounds__(256) void scl_patch_kernel(
    const float* __restrict__ img,      // [1,1,8192,8192]
    const float* __restrict__ tr,       // [1,256,256]
    float* __restrict__ partials)       // [gridDim.x][4]
{
  const int tid  = threadIdx.x;
  const int lane = tid & 31;
  const int wid  = tid >> 5;                     // wave in block: 0..7
  const int patch = blockIdx.x * 8 + wid;        // 0..65535
  const int py = patch >> 8;
  const int px = patch & 255;
  const float* base = img + (size_t)py * 32 * IMG_W + (size_t)px * 32;

  // Lane -> (row-within-4-row-group, column) so each b128 load covers
  // 4 complete 128B patch rows with zero waste.
  const int q    = lane & 15;
  const int hi   = lane >> 4;
  const int rsel = hi * 2 + (q >> 3);            // 0..3
  const int col  = (q & 7) * 4;                  // 0,4,...,28
  const float* lp = base + (size_t)rsel * IMG_W + col;

  v8f c = {0.f, 0.f, 0.f, 0.f, 0.f, 0.f, 0.f, 0.f};
  v2f ones; ones.x = 1.0f; ones.y = 1.0f;        // B = all-ones 4x16

#pragma unroll
  for (int j = 0; j < 8; ++j) {                  // 8 steps x 4 rows = 32 rows
    // Non-temporal: 256 MB single-pass stream, do not pollute WGP$/L2.
    v4f v = __builtin_nontemporal_load((const v4f*)(lp + (size_t)(4 * j) * IMG_W));
    v2f a0; a0.x = v.x; a0.y = v.y;
    v2f a1; a1.x = v.z; a1.y = v.w;
    // D = A*ones + C : every WMMA folds 64 patch elements into the f32 acc.
    c = __builtin_amdgcn_wmma_f32_16x16x4_f32(false, a0, false, ones,
                                              (short)0, c, false, false);
    c = __builtin_amdgcn_wmma_f32_16x16x4_f32(false, a1, false, ones,
                                              (short)0, c, false, false);
  }

  // D columns are identical (B constant): per-lane sum of 8 VGPRs gives
  // rows 0-7 (lanes 0-15) / rows 8-15 (lanes 16-31); one xor-16 finishes.
  float s = c[0] + c[1] + c[2] + c[3] + c[4] + c[5] + c[6] + c[7];
  s += __shfl_xor(s, 16, 32);

  __shared__ float red[8][4];
  if (lane == 0) {
    float mean  = s * (1.0f / 1024.0f);
    float maskf = (tr[patch] >= 0.5f) ? 1.0f : 0.0f;
    float z      = (mean - 0.34f) * 25.0f;              // / STD
    float logpdf = -0.5f * z * z - LOGC;
    float posv   = -__logf(__expf(logpdf) + 1e-6f);
    float negv   = mean * mean;
    red[wid][0] = posv * maskf;
    red[wid][1] = negv * (1.0f - maskf);
    red[wid][2] = maskf;
    red[wid][3] = 1.0f - maskf;
  }
  __syncthreads();
  if (tid < 4) {
    float acc = 0.f;
#pragma unroll
    for (int w = 0; w < 8; ++w) acc += red[w][tid];
    partials[(size_t)blockIdx.x * 4 + tid] = acc;
  }
}

__global__ __launch_bounds__(256) void scl_finalize_kernel(
    const float* __restrict__ partials,  // [8192][4]
    float* __restrict__ out)             // [1]
{
  __shared__ float r0[256], r1[256], r2[256], r3[256];
  const int tid = threadIdx.x;
  float a0 = 0.f, a1 = 0.f, a2 = 0.f, a3 = 0.f;
  const v4f* p4 = (const v4f*)partials;
  for (int g = tid; g < 8192; g += 256) {
    v4f v = p4[g];
    a0 += v.x; a1 += v.y; a2 += v.z; a3 += v.w;
  }
  r0[tid] = a0; r1[tid] = a1; r2[tid] = a2; r3[tid] = a3;
  __syncthreads();
  for (int off = 128; off > 0; off >>= 1) {
    if (tid < off) {
      r0[tid] += r0[tid + off];
      r1[tid] += r1[tid + off];
      r2[tid] += r2[tid + off];
      r3[tid] += r3[tid + off];
    }
    __syncthreads();
  }
  if (tid == 0)
    out[0] = r0[0] / r2[0] + r1[0] / r3[0];
}

extern "C" void kernel_launch(void* const* d_in, const int* in_sizes, int n_in,
                              void* d_out, int out_size, void* d_ws, size_t ws_size,
                              hipStream_t stream) {
  (void)in_sizes; (void)n_in; (void)out_size; (void)ws_size;
  const float* img = (const float*)d_in[0];   // unet_output, 8192*8192 fp32
  const float* tr  = (const float*)d_in[1];   // transformer_output, 256*256 fp32
  float* partials  = (float*)d_ws;            // 8192*4 floats = 128 KB scratch
  float* out       = (float*)d_out;           // scalar fp32

  scl_patch_kernel<<<8192, 256, 0, stream>>>(img, tr, partials);
  scl_finalize_kernel<<<1, 256, 0, stream>>>(partials, out);
}